// Izi_66692252172608
// MI455X (gfx1250) — compile-verified
//
#include <hip/hip_runtime.h>
#include <cstdint>

// Izhikevich recurrence, T=512 steps x N=65536 neurons, fp32.
// Bandwidth-bound (256 MB @ 23.3 TB/s ~= 11 us). CDNA5 path: deep async
// global->LDS pipeline (ASYNCcnt, GVS addressing, NT loads) + NT spike stores.

typedef float v4f __attribute__((ext_vector_type(4)));

#define IZI_T       512
#define IZI_N       65536
#define IZI_THREADS 128
#define IZI_VEC     4
#define IZI_DEPTH   16   // async pipeline depth (LDS ring slots)

#define IZI_A   0.02f
#define IZI_B   0.2f
#define IZI_C   (-0.065f)
#define IZI_D   0.008f
#define IZI_DT  (1.0f / 512.0f)
#define IZI_TH  0.3f

// Async DMA, GVS mode: copies 16B/lane from the global address (SGPR-pair base plus VGPR offset) to the LDS address in vdst.
// Uniform (time-varying) part of the address rides in the SGPR pair; the
// lane-varying neuron offset is a fixed 32-bit VGPR. Tracked by ASYNCcnt,
// retires in order. th:TH_LOAD_NT: x is a read-once 128 MB stream.
__device__ __forceinline__ void async_copy_b128(uint32_t lds_addr,
                                                uint32_t lane_byte_off,
                                                const void* sbase) {
  asm volatile("global_load_async_to_lds_b128 %0, %1, %2 th:TH_LOAD_NT"
               :
               : "v"(lds_addr), "v"(lane_byte_off),
                 "s"((unsigned long long)(uintptr_t)sbase)
               : "memory");
}

__device__ __forceinline__ void wait_async_15() {
  asm volatile("s_wait_asynccnt 15" ::: "memory");
}
__device__ __forceinline__ void wait_async_0() {
  asm volatile("s_wait_asynccnt 0" ::: "memory");
}

// One Izhikevich step for one neuron, op order mirrors the reference.
__device__ __forceinline__ void izi_step(float& v, float& r, float xt,
                                         float& spk) {
  v = (4.0f * v * v + 5.0f * v + 1.4f - r + xt) * IZI_DT;
  r = IZI_A * (IZI_B * v - v) * IZI_DT;
  const bool fire = (v >= IZI_TH);
  spk = fire ? 1.0f : 0.0f;
  v   = fire ? IZI_C : v;
  r   = fire ? (r + IZI_D) : r;
}

__global__ __launch_bounds__(IZI_THREADS) void Izi_66692252172608_kernel(
    const float* __restrict__ x, float* __restrict__ out) {
  // LDS ring buffer: DEPTH slots x (THREADS * 16B) = 32 KB per workgroup.
  __shared__ v4f buf[IZI_DEPTH][IZI_THREADS];

  const int tid  = threadIdx.x;
  const int base = blockIdx.x * (IZI_THREADS * IZI_VEC) + tid * IZI_VEC;

  const uint32_t lane_off = (uint32_t)base * 4u;   // byte offset within a row
  const char*    sx       = (const char*)x;        // scalar base, stepped by t
  float*         gout     = out + base;

  // LDS byte address of this lane's slot 0 element (low 32 bits of the flat
  // address are the LDS offset on gfx1250).
  const uint32_t lds0   = (uint32_t)(uintptr_t)(&buf[0][tid]);
  const uint32_t lslice = (uint32_t)(sizeof(v4f) * IZI_THREADS);  // 2048 B

  // ---- prologue: fill the pipeline with DEPTH async copies -------------
#pragma unroll
  for (int i = 0; i < IZI_DEPTH; ++i) {
    async_copy_b128(lds0 + (uint32_t)i * lslice, lane_off,
                    sx + (size_t)i * IZI_N * 4);
  }

  float v[IZI_VEC], r[IZI_VEC];
#pragma unroll
  for (int j = 0; j < IZI_VEC; ++j) {
    v[j] = IZI_C;
    r[j] = 0.0f;
  }

  // ---- steady state ----------------------------------------------------
  for (int t = 0; t < IZI_T - IZI_DEPTH; ++t) {
    // DEPTH copies outstanding (t .. t+DEPTH-1); <=15 left => slot t done.
    wait_async_15();
    const int slot = t & (IZI_DEPTH - 1);

    const v4f xv = buf[slot][tid];  // ds_load_b128, own bytes only

    v4f s;
#pragma unroll
    for (int j = 0; j < IZI_VEC; ++j) {
      float sp;
      izi_step(v[j], r[j], xv[j], sp);
      s[j] = sp;
    }

    // Streamed once: non-temporal 128-bit store.
    __builtin_nontemporal_store(s, (v4f*)(gout + (size_t)t * IZI_N));

    // Refill this slot with timestep t+DEPTH (uniform base advance -> SALU).
    async_copy_b128(lds0 + (uint32_t)slot * lslice, lane_off,
                    sx + (size_t)(t + IZI_DEPTH) * IZI_N * 4);
  }

  // ---- drain tail ------------------------------------------------------
  wait_async_0();
#pragma unroll
  for (int t = IZI_T - IZI_DEPTH; t < IZI_T; ++t) {
    const int slot = t & (IZI_DEPTH - 1);
    const v4f xv = buf[slot][tid];

    v4f s;
#pragma unroll
    for (int j = 0; j < IZI_VEC; ++j) {
      float sp;
      izi_step(v[j], r[j], xv[j], sp);
      s[j] = sp;
    }
    __builtin_nontemporal_store(s, (v4f*)(gout + (size_t)t * IZI_N));
  }
}

extern "C" void kernel_launch(void* const* d_in, const int* in_sizes, int n_in,
                              void* d_out, int out_size, void* d_ws,
                              size_t ws_size, hipStream_t stream) {
  (void)in_sizes; (void)n_in; (void)out_size; (void)d_ws; (void)ws_size;
  const float* x = (const float*)d_in[0];
  float* out     = (float*)d_out;

  const int neurons_per_block = IZI_THREADS * IZI_VEC;        // 512
  const int blocks            = IZI_N / neurons_per_block;    // 128

  Izi_66692252172608_kernel<<<blocks, IZI_THREADS, 0, stream>>>(x, out);
}